// DirectedProcessVGAE_43722767073863
// MI455X (gfx1250) — compile-verified
//
#include <hip/hip_runtime.h>

#define NNODES 8192
#define NCH    128
#define NEDGE  262144
#define NC     (NNODES * NCH)   // 1,048,576 floats per node-feature buffer

typedef float v2f __attribute__((ext_vector_type(2)));
typedef float v8f __attribute__((ext_vector_type(8)));

// ---------------------------------------------------------------------------
// degree / normalization
// ---------------------------------------------------------------------------
__global__ __launch_bounds__(256) void deg_init_kernel(float* deg) {
    int i = blockIdx.x * blockDim.x + threadIdx.x;
    if (i < NNODES) deg[i] = 1.0f;   // self-loop
}

__global__ __launch_bounds__(256) void deg_scatter_kernel(const long long* __restrict__ ei,
                                                          float* __restrict__ deg) {
    int e = blockIdx.x * blockDim.x + threadIdx.x;
    if (e < NEDGE) {
        int dst = (int)ei[NEDGE + e];
        unsafeAtomicAdd(&deg[dst], 1.0f);   // global_atomic_add_f32
    }
}

__global__ __launch_bounds__(256) void dis_kernel(const float* __restrict__ deg,
                                                  float* __restrict__ dis) {
    int i = blockIdx.x * blockDim.x + threadIdx.x;
    if (i < NNODES) dis[i] = rsqrtf(deg[i]);
}

// ---------------------------------------------------------------------------
// Simple fp32 WMMA GEMM for the per-layer dense step: D[M,N] = A[M,K] @ B[K,N]
// One wave per 16x16 tile, K-loop of V_WMMA_F32_16X16X4_F32.
// ---------------------------------------------------------------------------
template <int M, int N, int K>
__global__ __launch_bounds__(256) void wmma_gemm_f32(const float* __restrict__ A,
                                                     const float* __restrict__ B,
                                                     float* __restrict__ D) {
    const int lane = threadIdx.x & 31;
    const int wave = threadIdx.x >> 5;
    const int tile = blockIdx.x * 8 + wave;
    const int tilesN = N / 16;
    const int tm = tile / tilesN;
    const int tn = tile % tilesN;
    if (tm >= M / 16) return;            // wave-uniform guard (EXEC stays all-1)

    const int half = lane >> 4;          // 0: lanes 0-15, 1: lanes 16-31
    const int idx  = lane & 15;
    const int row  = tm * 16 + idx;      // A row for this lane
    const int col  = tn * 16 + idx;      // B column / D column for this lane

    v8f c = {};
    for (int k0 = 0; k0 < K; k0 += 4) {
        const int ka = k0 + 2 * half;    // this lane's k pair: {ka, ka+1}
        v2f a = *(const v2f*)(A + row * K + ka);
        v2f b;
        b.x = B[ka * N + col];
        b.y = B[(ka + 1) * N + col];
        c = __builtin_amdgcn_wmma_f32_16x16x4_f32(
            false, a, false, b, (short)0, c, false, false);
    }

#pragma unroll
    for (int r = 0; r < 8; ++r) {
        D[(tm * 16 + r + 8 * half) * N + col] = c[r];
    }
}

// ---------------------------------------------------------------------------
// Register-blocked fp32 WMMA GEMM with transposed B:  D[M,N] = A[M,K] @ B^T,
// B stored [N,K].  Each wave owns a (16*BI) x (16*BJ) output block:
// per k-step, BI+BJ float2 loads feed BI*BJ WMMAs (outer product), raising
// L2 arithmetic intensity by BI*BJ/((BI+BJ)/2)x vs the naive 1-tile wave.
// BI=BJ=4: 64 KB of L2 reads per 1.05 MFLOP -> 16 flop/byte, compute-bound.
// ---------------------------------------------------------------------------
template <int M, int N, int K, int BI, int BJ>
__global__ __launch_bounds__(256) void wmma_gemm_f32_bt_blk(const float* __restrict__ A,
                                                            const float* __restrict__ B,
                                                            float* __restrict__ D) {
    const int lane = threadIdx.x & 31;
    const int wave = threadIdx.x >> 5;
    const int tile = blockIdx.x * 8 + wave;
    const int tilesN = N / (16 * BJ);
    const int tm = tile / tilesN;
    const int tn = tile % tilesN;
    if (tm >= M / (16 * BI)) return;     // wave-uniform guard

    const int half = lane >> 4;
    const int idx  = lane & 15;

    const float* Ab = A + ((size_t)(tm * 16 * BI) + idx) * K;  // lane's A rows
    const float* Bb = B + ((size_t)(tn * 16 * BJ) + idx) * K;  // lane's B rows

    v8f acc[BI][BJ] = {};

    for (int k0 = 0; k0 < K; k0 += 4) {
        const int ka = k0 + 2 * half;
        v2f a[BI], b[BJ];
#pragma unroll
        for (int i = 0; i < BI; ++i) a[i] = *(const v2f*)(Ab + (size_t)(i * 16) * K + ka);
#pragma unroll
        for (int j = 0; j < BJ; ++j) b[j] = *(const v2f*)(Bb + (size_t)(j * 16) * K + ka);
#pragma unroll
        for (int i = 0; i < BI; ++i)
#pragma unroll
            for (int j = 0; j < BJ; ++j)
                acc[i][j] = __builtin_amdgcn_wmma_f32_16x16x4_f32(
                    false, a[i], false, b[j], (short)0, acc[i][j], false, false);
    }

#pragma unroll
    for (int i = 0; i < BI; ++i)
#pragma unroll
        for (int j = 0; j < BJ; ++j)
#pragma unroll
            for (int r = 0; r < 8; ++r)
                D[((size_t)(tm * 16 * BI + i * 16 + r + 8 * half)) * N +
                  tn * 16 * BJ + j * 16 + idx] = acc[i][j][r];
}

// ---------------------------------------------------------------------------
// GCN aggregation: self-loop + bias init, then edge scatter with fp32 atomics
// ---------------------------------------------------------------------------
__global__ __launch_bounds__(256) void gcn_init_kernel(const float* __restrict__ h,
                                                       const float* __restrict__ dis,
                                                       const float* __restrict__ bias,
                                                       float* __restrict__ agg) {
    int i = blockIdx.x * blockDim.x + threadIdx.x;   // over NNODES*NCH
    if (i < NC) {
        int node = i >> 7;       // / 128
        int ch   = i & 127;
        float d  = dis[node];
        agg[i] = h[i] * d * d + bias[ch];
    }
}

// one wave per edge: 32 lanes x 4 channels (float4 gather, 4 fp32 atomics)
__global__ __launch_bounds__(256) void gcn_scatter_kernel(const float* __restrict__ h,
                                                          const long long* __restrict__ ei,
                                                          const float* __restrict__ dis,
                                                          float* __restrict__ agg) {
    int gid = blockIdx.x * blockDim.x + threadIdx.x;
    int e = gid >> 5;
    int c = (gid & 31) << 2;
    if (e < NEDGE) {
        int src = (int)ei[e];
        int dst = (int)ei[NEDGE + e];
        float nrm = dis[src] * dis[dst];
        const float4 h4 = *(const float4*)(h + src * NCH + c);
        float* out = agg + dst * NCH + c;
        unsafeAtomicAdd(out + 0, h4.x * nrm);
        unsafeAtomicAdd(out + 1, h4.y * nrm);
        unsafeAtomicAdd(out + 2, h4.z * nrm);
        unsafeAtomicAdd(out + 3, h4.w * nrm);
    }
}

__global__ __launch_bounds__(256) void relu_kernel(float* __restrict__ x, int n) {
    int i = blockIdx.x * blockDim.x + threadIdx.x;
    if (i < n) x[i] = fmaxf(x[i], 0.0f);
}

// ---------------------------------------------------------------------------
// launch
// ---------------------------------------------------------------------------
extern "C" void kernel_launch(void* const* d_in, const int* in_sizes, int n_in,
                              void* d_out, int out_size, void* d_ws, size_t ws_size,
                              hipStream_t stream) {
    const float*     x  = (const float*)d_in[0];
    const long long* ei = (const long long*)d_in[1];
    const float* Ws  = (const float*)d_in[2];  const float* bs  = (const float*)d_in[3];
    const float* Wt  = (const float*)d_in[4];  const float* bt  = (const float*)d_in[5];
    const float* W1  = (const float*)d_in[6];  const float* b1  = (const float*)d_in[7];
    const float* W2  = (const float*)d_in[8];  const float* b2  = (const float*)d_in[9];
    const float* Wmu = (const float*)d_in[10]; const float* bmu = (const float*)d_in[11];
    const float* W5  = (const float*)d_in[12]; const float* b5  = (const float*)d_in[13];
    const float* W6  = (const float*)d_in[14]; const float* b6  = (const float*)d_in[15];

    float* adj  = (float*)d_out;                     // [8192, 8192]
    float* hout = adj + (size_t)NNODES * NNODES;     // [8192, 128]

    float* ws  = (float*)d_ws;
    float* deg = ws;                 // 8192
    float* dis = deg + NNODES;       // 8192
    float* tmp = dis + NNODES;       // NC (dense GEMM result per layer)
    float* sS  = tmp + NC;           // NC
    float* sT  = sS + NC;            // NC
    float* hA  = sT + NC;            // NC
    float* hB  = hA + NC;            // NC

    // ---- degree / normalization (once, shared by all layers) ----
    deg_init_kernel<<<NNODES / 256, 256, 0, stream>>>(deg);
    deg_scatter_kernel<<<NEDGE / 256, 256, 0, stream>>>(ei, deg);
    dis_kernel<<<NNODES / 256, 256, 0, stream>>>(deg, dis);

    const int denseBlocks  = (NNODES / 16) * (NCH / 16) / 8;   // 512
    const int elemBlocks   = NC / 256;                         // 4096
    const int edgeBlocks   = (NEDGE * 32) / 256;               // 32768

    auto run_gcn = [&](const float* xin, const float* W, const float* b,
                       float* agg, bool relu) {
        wmma_gemm_f32<NNODES, NCH, NCH><<<denseBlocks, 256, 0, stream>>>(xin, W, tmp);
        gcn_init_kernel<<<elemBlocks, 256, 0, stream>>>(tmp, dis, b, agg);
        gcn_scatter_kernel<<<edgeBlocks, 256, 0, stream>>>(tmp, ei, dis, agg);
        if (relu) relu_kernel<<<elemBlocks, 256, 0, stream>>>(agg, NC);
    };

    // ---- edge encoder ----
    run_gcn(x, Ws, bs, sS, false);
    run_gcn(x, Wt, bt, sT, false);

    // ---- NodeModel chain ----
    run_gcn(x,  W1,  b1,  hA,   true);
    run_gcn(hA, W2,  b2,  hB,   true);
    run_gcn(hB, Wmu, bmu, hA,   false);
    run_gcn(hA, W5,  b5,  hB,   true);
    run_gcn(hB, W6,  b6,  hout, true);

    // ---- dominant op: adj = s @ t.T, 8192x8192x128 fp32, 64x64 per wave ----
    const int adjBlocks = (NNODES / 64) * (NNODES / 64) / 8;   // 2048
    wmma_gemm_f32_bt_blk<NNODES, NNODES, NCH, 4, 4>
        <<<adjBlocks, 256, 0, stream>>>(sS, sT, adj);
}